// PointLoss_77532749628013
// MI455X (gfx1250) — compile-verified
//
#include <hip/hip_runtime.h>
#include <hip/hip_bf16.h>
#include <stdint.h>

#define EPS     1e-7f
#define TRUNC_  0.5f
#define NROW    256
#define NPTS    2048
#define NCOL    6144            // 3*NPTS, flattened row length
#define KCAND   64
#define THREADS 256
#define NC      (NCOL / THREADS)   // 24 candidates per thread (extrema kernel)
#define TILE    1024
#define NTILES  (NCOL / TILE)      // 6

typedef __attribute__((ext_vector_type(2))) float v2f;
typedef __attribute__((ext_vector_type(8))) float v8f;

// ---------- CDNA5 async global->LDS helpers (gfx1250) ----------
__device__ __forceinline__ unsigned lds_byte_off(void* p) {
    // generic -> LDS(addrspace 3); AS3 pointers are 32-bit LDS offsets
    return (unsigned)(uintptr_t)((__attribute__((address_space(3))) void*)p);
}
__device__ __forceinline__ void async_load_b128(void* lds_dst, const void* gsrc) {
    asm volatile("global_load_async_to_lds_b128 %0, %1, off"
                 :: "v"(lds_byte_off(lds_dst)), "v"(gsrc)
                 : "memory");
}
template <int N>
__device__ __forceinline__ void async_wait() {
    asm volatile("s_wait_asynccnt %0" :: "n"(N) : "memory");
}

// ---------------------------------------------------------------
// Kernel 1: derived arrays (sign-normalized), f32.
//   ydx = ys/max(xs,EPS), wx = w*xs, wyn = -(w*ys),
//   cb = (wy-T)/max(wx,EPS), cc = (wy+T)/max(wx,EPS)
// ---------------------------------------------------------------
__global__ void prep_kernel(const float* __restrict__ src,
                            const float* __restrict__ tgt,
                            const float* __restrict__ wgt,
                            float* __restrict__ YDX, float* __restrict__ WX,
                            float* __restrict__ WYN, float* __restrict__ CB,
                            float* __restrict__ CC) {
    const int total = NROW * NCOL;
    for (int gid = blockIdx.x * blockDim.x + threadIdx.x; gid < total;
         gid += gridDim.x * blockDim.x) {
        int row = gid / NCOL;
        int col = gid - row * NCOL;
        float xo = src[gid], yo = tgt[gid];
        float s  = (xo > 0.f) ? 1.f : ((xo < 0.f) ? -1.f : 0.f);
        float xs = xo * s, ys = yo * s;
        float w  = wgt[row * NPTS + col / 3];
        float wxv = w * xs, wyv = w * ys;
        float mwx = fmaxf(wxv, EPS);
        YDX[gid] = ys / fmaxf(xs, EPS);
        WX[gid]  = wxv;
        WYN[gid] = -wyv;
        CB[gid]  = (wyv - TRUNC_) / mwx;
        CC[gid]  = (wyv + TRUNC_) / mwx;
    }
}

// ---------------------------------------------------------------
// Kernel 2: extrema detection (replaces sort+searchsorted).
//   left_d(a)  = 2*S<(ydx,a) - S<(cb,a) - S<(cc,a)   (S = wx-weighted count)
//   right_d(a) = same with <=
//   extremum iff left_d < 0 && right_d >= 0; keep first 64 by index.
// One block per row; double-buffered async global->LDS streaming.
// ---------------------------------------------------------------
__global__ __launch_bounds__(THREADS) void extrema_kernel(
        const float* __restrict__ YDX, const float* __restrict__ WX,
        const float* __restrict__ CB,  const float* __restrict__ CC,
        int* __restrict__ cand_idx, int* __restrict__ cand_cnt) {
    __shared__ __align__(16) float s_y[2][TILE], s_wx[2][TILE],
                                   s_cb[2][TILE], s_cc[2][TILE];
    __shared__ unsigned s_flags[NCOL / 32];   // 192 words

    const int r = blockIdx.x, t = threadIdx.x;
    const float* ydx_row = YDX + (size_t)r * NCOL;
    const float* wx_row  = WX  + (size_t)r * NCOL;
    const float* cb_row  = CB  + (size_t)r * NCOL;
    const float* cc_row  = CC  + (size_t)r * NCOL;

#define ISSUE_TILE(buf, t0)                                          \
    do {                                                             \
        async_load_b128(&s_y [buf][4 * t], ydx_row + (t0) + 4 * t);  \
        async_load_b128(&s_wx[buf][4 * t], wx_row  + (t0) + 4 * t);  \
        async_load_b128(&s_cb[buf][4 * t], cb_row  + (t0) + 4 * t);  \
        async_load_b128(&s_cc[buf][4 * t], cc_row  + (t0) + 4 * t);  \
    } while (0)

    ISSUE_TILE(0, 0);                          // prefetch first tile

    float ca[NC], dl[NC], dr[NC];
#pragma unroll
    for (int c = 0; c < NC; ++c) {
        ca[c] = ydx_row[t + THREADS * c];
        dl[c] = 0.f; dr[c] = 0.f;
    }
    for (int w = t; w < NCOL / 32; w += THREADS) s_flags[w] = 0u;

#define DERIV_BODY(yj, wxj, cbj, ccj)                                        \
    do {                                                                     \
        _Pragma("unroll")                                                    \
        for (int c = 0; c < NC; ++c) {                                       \
            float a  = ca[c];                                                \
            float sl = ((yj) <  a ? 2.f : 0.f) - ((cbj) <  a ? 1.f : 0.f)    \
                                              - ((ccj) <  a ? 1.f : 0.f);    \
            float sr = ((yj) <= a ? 2.f : 0.f) - ((cbj) <= a ? 1.f : 0.f)    \
                                              - ((ccj) <= a ? 1.f : 0.f);    \
            dl[c] = fmaf((wxj), sl, dl[c]);                                  \
            dr[c] = fmaf((wxj), sr, dr[c]);                                  \
        }                                                                    \
    } while (0)

    for (int ti = 0; ti < NTILES; ++ti) {
        const int cur = ti & 1;
        if (ti + 1 < NTILES) {
            // buffer cur^1 was released by the trailing barrier of ti-1
            ISSUE_TILE(cur ^ 1, (ti + 1) * TILE);
            async_wait<4>();                   // newest 4 (next tile) may remain
        } else {
            async_wait<0>();
        }
        __syncthreads();                       // tile `cur` visible to all waves
        const float4* y4  = (const float4*)s_y [cur];
        const float4* wx4 = (const float4*)s_wx[cur];
        const float4* cb4 = (const float4*)s_cb[cur];
        const float4* cc4 = (const float4*)s_cc[cur];
        for (int j4 = 0; j4 < TILE / 4; ++j4) {
            float4 yv = y4[j4], wv = wx4[j4], bv = cb4[j4], cv = cc4[j4];
            DERIV_BODY(yv.x, wv.x, bv.x, cv.x);
            DERIV_BODY(yv.y, wv.y, bv.y, cv.y);
            DERIV_BODY(yv.z, wv.z, bv.z, cv.z);
            DERIV_BODY(yv.w, wv.w, bv.w, cv.w);
        }
        __syncthreads();                       // release buffer `cur`
    }
#undef DERIV_BODY
#undef ISSUE_TILE

#pragma unroll
    for (int c = 0; c < NC; ++c) {
        if (dl[c] < 0.f && dr[c] >= 0.f) {
            int i = t + THREADS * c;
            atomicOr(&s_flags[i >> 5], 1u << (i & 31));
        }
    }
    __syncthreads();
    if (t == 0) {                              // ordered collection of first 64
        int e = 0;
        for (int w = 0; w < NCOL / 32 && e < KCAND; ++w) {
            unsigned m = s_flags[w];
            while (m && e < KCAND) {
                int b = __ffs(m) - 1;
                cand_idx[r * KCAND + e] = (w << 5) | b;
                ++e;
                m &= m - 1;
            }
        }
        if (e == 0) { cand_idx[r * KCAND] = 0; e = 1; }  // reference fallback
        cand_cnt[r] = e;
    }
}

// ---------------------------------------------------------------
// Kernel 3: residual scan over <=64 candidates via V_WMMA_F32_16X16X4_F32.
// A uses columns K=0 (lanes 0-15) and K=2 (lanes 16-31) of VGPR A.x:
//   A.x = lo ? a_M : 1.0 ; A.y = 0
// B VGPR B.x holds rows K=0 (lanes 0-15) and K=2 (lanes 16-31):
//   B.x = lo ? wx[j] : (-wy)[j] ; B.y = 0     (base-pointer select, no branch)
// => D[M][N] = a_M*wx_j - wy_j ; VALU: acc += min(|D|, TRUNC); argmin.
// Wave w: candidate tile (w&3), j-half (w>>2). One block per row.
// ---------------------------------------------------------------
__global__ __launch_bounds__(THREADS) void residual_kernel(
        const float* __restrict__ YDX, const float* __restrict__ WX,
        const float* __restrict__ WYN,
        const int* __restrict__ cand_idx, const int* __restrict__ cand_cnt,
        float* __restrict__ out) {
    __shared__ __align__(16) float s_wx[NCOL];      // 24 KB
    __shared__ __align__(16) float s_wyn[NCOL];     // 24 KB (already negated)
    __shared__ float s_a[KCAND];
    __shared__ float s_part[8 * 32 * 8];            // 8 KB wave partials
    __shared__ float s_res[KCAND];

    const int r = blockIdx.x, t = threadIdx.x;
    const float* wx_row  = WX  + (size_t)r * NCOL;
    const float* wyn_row = WYN + (size_t)r * NCOL;
#pragma unroll
    for (int q = 0; q < NCOL / (4 * THREADS); ++q) {   // 6 chunks of b128
        int e0 = 4 * (t + THREADS * q);
        async_load_b128(&s_wx[e0],  wx_row  + e0);
        async_load_b128(&s_wyn[e0], wyn_row + e0);
    }
    const int E = cand_cnt[r];
    if (t < KCAND) {
        float a = 0.f;
        if (t < E) a = YDX[(size_t)r * NCOL + cand_idx[r * KCAND + t]];
        s_a[t] = a;
    }
    async_wait<0>();
    __syncthreads();

    const int wave = t >> 5, lane = t & 31;
    const int ktile = wave & 3, jhalf = wave >> 2;
    const int l15 = lane & 15;
    const bool lo = lane < 16;
    const float* base = lo ? s_wx : s_wyn;     // hoisted lane-half select

    v2f A;
    A.x = lo ? s_a[ktile * 16 + l15] : 1.0f;   // K=0: a_M | K=2: 1.0
    A.y = 0.f;                                 // K=1, K=3 columns are zero
    v8f acc = {0.f, 0.f, 0.f, 0.f, 0.f, 0.f, 0.f, 0.f};
    v8f cz  = {0.f, 0.f, 0.f, 0.f, 0.f, 0.f, 0.f, 0.f};

    const int jbase = jhalf * (NCOL / 2) + l15;
#pragma unroll 2
    for (int jt = 0; jt < (NCOL / 2) / 16; ++jt) {      // 192 WMMA per wave
        v2f Bm;
        Bm.x = base[jbase + jt * 16];          // K=0: wx | K=2: -wy
        Bm.y = 0.f;                            // K=1, K=3 rows are zero
        v8f d = __builtin_amdgcn_wmma_f32_16x16x4_f32(
                    false, A, false, Bm, (short)0, cz, false, false);
#pragma unroll
        for (int v = 0; v < 8; ++v)
            acc[v] += fminf(fabsf(d[v]), TRUNC_);
    }
#pragma unroll
    for (int v = 0; v < 8; ++v)
        s_part[(wave * 32 + lane) * 8 + v] = acc[v];
    __syncthreads();

    if (t < KCAND) {   // D layout: VGPR v holds M=v (lanes 0-15), M=v+8 (16-31)
        int kt = t >> 4, m = t & 15, v = m & 7, half = m >> 3;
        float sum = 0.f;
        for (int h = 0; h < 2; ++h) {
            int w = kt + 4 * h;
            int lbase = half ? 16 : 0;
            for (int l = 0; l < 16; ++l)
                sum += s_part[(w * 32 + lbase + l) * 8 + v];
        }
        s_res[t] = (t < E) ? sum : __builtin_inff();
    }
    __syncthreads();
    if (t == 0) {      // first-occurrence argmin, like jnp.argmin
        float best = s_res[0];
        int kb = 0;
        for (int k = 1; k < KCAND; ++k)
            if (s_res[k] < best) { best = s_res[k]; kb = k; }
        out[r] = s_a[kb];
    }
}

// ---------------------------------------------------------------
extern "C" void kernel_launch(void* const* d_in, const int* in_sizes, int n_in,
                              void* d_out, int out_size, void* d_ws, size_t ws_size,
                              hipStream_t stream) {
    (void)in_sizes; (void)n_in; (void)out_size; (void)ws_size;
    const float* src = (const float*)d_in[0];   // (256,2048,3) f32
    const float* tgt = (const float*)d_in[1];   // (256,2048,3) f32
    const float* wgt = (const float*)d_in[2];   // (256,2048)   f32
    float* out = (float*)d_out;                 // (256,)       f32

    const size_t R = (size_t)NROW * NCOL;
    float* YDX = (float*)d_ws;                  // ws usage: 5*R floats + cands
    float* WX  = YDX + R;
    float* WYN = WX + R;
    float* CB  = WYN + R;
    float* CC  = CB + R;
    int*   cidx = (int*)(CC + R);
    int*   ccnt = cidx + NROW * KCAND;

    prep_kernel<<<256, THREADS, 0, stream>>>(src, tgt, wgt, YDX, WX, WYN, CB, CC);
    extrema_kernel<<<NROW, THREADS, 0, stream>>>(YDX, WX, CB, CC, cidx, ccnt);
    residual_kernel<<<NROW, THREADS, 0, stream>>>(YDX, WX, WYN, cidx, ccnt, out);
}